// SeqOperation_28106265985064
// MI455X (gfx1250) — compile-verified
//
#include <hip/hip_runtime.h>

#ifndef __has_builtin
#define __has_builtin(x) 0
#endif

// Problem shape from the reference: seq_emb (B=4096, n=200, d=128) fp32,
// pos_idx (B,) int. Pure data movement (bandwidth-bound, ~838 MB total):
//   out[b,j] = valid*E[b,j]*(j<p) + E[b,p]*(j==(p+1)%n) + valid*E[b,j-2]*(j>p+2)
// with p = pos_idx[b], valid = (p <= n-3).

namespace {
constexpr int kN          = 200;  // sequence length
constexpr int kD          = 128;  // embedding dim (= 32 lanes * float4)
constexpr int kRowsPerBlk = 64;   // output rows per block
constexpr int kBlock      = 256;  // 8 wave32 waves
constexpr int kChunks     = (kN + kRowsPerBlk - 1) / kRowsPerBlk;  // 4
constexpr int kTileRows   = kRowsPerBlk + 2;                       // 66 (rows j and j-2)
}  // namespace

typedef float v4f __attribute__((ext_vector_type(4)));
typedef int   v4i __attribute__((ext_vector_type(4)));
typedef __attribute__((address_space(1))) v4i* as1_v4i;  // global
typedef __attribute__((address_space(3))) v4i* as3_v4i;  // LDS

__global__ __launch_bounds__(kBlock) void seqop_shift_gather_kernel(
    const float* __restrict__ seq, const int* __restrict__ pos_idx,
    float* __restrict__ out) {
  __shared__ float tile[kTileRows * kD];  // 33,792 bytes

  const int chunk = blockIdx.x;
  const int b     = blockIdx.y;
  const int tid   = threadIdx.x;

  const int base = chunk * kRowsPerBlk;
  const int lo   = (base >= 2) ? (base - 2) : 0;
  const int hi   = (base + kRowsPerBlk < kN) ? (base + kRowsPerBlk) : kN;

  const float* __restrict__ src = seq + (size_t)b * (kN * kD);
  float* __restrict__ dst       = out + (size_t)b * (kN * kD);

  // ---------------- Stage rows [lo, hi) into LDS ----------------
  const int nvec = (hi - lo) * (kD / 4);  // #float4 elements, <= 2112
#if defined(__gfx1250__) && __has_builtin(__builtin_amdgcn_global_load_async_to_lds_b128)
  {
    const char* g = (const char*)(src + (size_t)lo * kD);
    char*       s = (char*)tile;
    for (int i = tid; i < nvec; i += kBlock) {
      __builtin_amdgcn_global_load_async_to_lds_b128(
          (as1_v4i)(g + (size_t)i * 16), (as3_v4i)(s + (size_t)i * 16),
          /*offset=*/0, /*cpol=*/0);
    }
  }
#if __has_builtin(__builtin_amdgcn_s_wait_asynccnt)
  __builtin_amdgcn_s_wait_asynccnt(0);
#else
  asm volatile("s_wait_asynccnt 0" ::: "memory");
#endif
#else
  {
    const v4f* g4 = (const v4f*)(src + (size_t)lo * kD);
    v4f*       s4 = (v4f*)tile;
    for (int i = tid; i < nvec; i += kBlock) s4[i] = g4[i];
  }
#endif
  __syncthreads();

  // ---------------- Compute & stream out ----------------
  const int  p     = pos_idx[b];
  const bool valid = (p <= kN - 3);
  const int  col   = (tid & 31) * 4;  // float offset within a row
  const int  rgrp  = tid >> 5;        // 0..7: row group within block
  const int  jr1   = (p + 1 < kN) ? (p + 1) : 0;  // roll-by-1 landing row

  // Row p for the roll-by-1 term: uniform address -> L0 hit for all waves.
  const v4f pv = *(const v4f*)(src + (size_t)p * kD + col);

  const int rows_out = hi - base;
  for (int jj = rgrp; jj < rows_out; jj += kBlock / 32) {
    const int j = base + jj;
    v4f acc = {0.f, 0.f, 0.f, 0.f};
    if (valid) {
      if (j < p) {
        acc = *(const v4f*)&tile[(j - lo) * kD + col];          // left term
      } else if (j > p + 2) {
        acc = *(const v4f*)&tile[(j - 2 - lo) * kD + col];      // rolled-right term
      }
    }
    if (j == jr1) acc += pv;                                    // rolled-pos term
    // Write-once stream: non-temporal b128 store.
    __builtin_nontemporal_store(acc, (v4f*)(dst + (size_t)j * kD + col));
  }
}

extern "C" void kernel_launch(void* const* d_in, const int* in_sizes, int n_in,
                              void* d_out, int out_size, void* d_ws, size_t ws_size,
                              hipStream_t stream) {
  (void)n_in; (void)out_size; (void)d_ws; (void)ws_size;
  const float* seq = (const float*)d_in[0];
  const int*   pos = (const int*)d_in[1];
  float*       out = (float*)d_out;
  const int B = in_sizes[1];  // 4096 batch elements (pos_idx count)
  dim3 grid(kChunks, B);
  seqop_shift_gather_kernel<<<grid, kBlock, 0, stream>>>(seq, pos, out);
}